// LawGraphEncoder_77515569758941
// MI455X (gfx1250) — compile-verified
//
#include <hip/hip_runtime.h>

typedef float v2f __attribute__((ext_vector_type(2)));
typedef float v8f __attribute__((ext_vector_type(8)));

// ---------------------------------------------------------------- utilities
__global__ void zero_f32(float* __restrict__ p, long long n) {
    long long i = (long long)blockIdx.x * blockDim.x + threadIdx.x;
    long long stride = (long long)gridDim.x * blockDim.x;
    for (; i < n; i += stride) p[i] = 0.0f;
}

__global__ void degree_kernel(const int* __restrict__ src, const int* __restrict__ dst,
                              float* __restrict__ outdeg, float* __restrict__ indeg, int E) {
    int e = blockIdx.x * blockDim.x + threadIdx.x;
    if (e < E) {
        unsafeAtomicAdd(&outdeg[src[e]], 1.0f);
        unsafeAtomicAdd(&indeg[dst[e]], 1.0f);
    }
}

__global__ void norm_kernel(const float* __restrict__ outdeg, const float* __restrict__ indeg,
                            float* __restrict__ out_norm, float* __restrict__ in_norm, int N) {
    int i = blockIdx.x * blockDim.x + threadIdx.x;
    if (i < N) {
        out_norm[i] = rsqrtf(fmaxf(outdeg[i], 1.0f));
        in_norm[i]  = rsqrtf(fmaxf(indeg[i], 1.0f));
    }
}

// ------------------------------------------------- fp32 WMMA GEMM: Y = diag(norm) * (X @ W)
// X: [N,K] row-major, W: [K,OUTC] row-major (staged in LDS), Y: [N,OUTC].
// The src-side degree norm is a per-row scalar, so it is folded into the
// accumulator at store time (GEMM is linear over rows) — keeps the K-loop
// as pure load -> v_wmma with no VALU on the critical path.
// 8 waves per block; each wave computes one 16x16 tile via V_WMMA_F32_16X16X4_F32.
template <int K, int OUTC>
__global__ void gemm_wmma(const float* __restrict__ X, const float* __restrict__ W,
                          const float* __restrict__ norm, float* __restrict__ Y, int N) {
    constexpr int COLT = OUTC / 16;   // column tiles covered by the block
    constexpr int ROWT = 8 / COLT;    // row tiles per block (8 waves total)
    constexpr int BM   = 16 * ROWT;

    __shared__ float Ws[K * OUTC];
    for (int i = threadIdx.x; i < K * OUTC; i += blockDim.x) Ws[i] = W[i];
    __syncthreads();

    const int wv      = threadIdx.x >> 5;
    const int lane    = threadIdx.x & 31;
    const int rowTile = wv / COLT;
    const int colTile = wv % COLT;

    const int m    = lane & 15;                 // A: row within tile / B: col within tile
    const int koff = (lane < 16) ? 0 : 2;       // ISA A/B fragment layout: half-wave K split

    const int row  = blockIdx.x * BM + rowTile * 16 + m;
    const int rowc = min(row, N - 1);           // clamp (EXEC must stay all-1s for WMMA)
    const float* xrow = X + (size_t)rowc * K;
    const int colbase = colTile * 16;

    v8f acc = {};
#pragma unroll
    for (int kk = 0; kk < K; kk += 4) {
        v2f a, b;
        a.x = xrow[kk + koff];                  // merged global_load_b64
        a.y = xrow[kk + koff + 1];
        b.x = Ws[(kk + koff)     * OUTC + colbase + m];
        b.y = Ws[(kk + koff + 1) * OUTC + colbase + m];
        acc = __builtin_amdgcn_wmma_f32_16x16x4_f32(
            /*neg_a=*/false, a, /*neg_b=*/false, b,
            /*c_mod=*/(short)0, acc, /*reuse_a=*/false, /*reuse_b=*/false);
    }

    // C/D layout: VGPR r -> M = r (lanes 0-15) / M = r+8 (lanes 16-31), N = lane&15
    const int rbase = blockIdx.x * BM + rowTile * 16 + ((lane < 16) ? 0 : 8);
#pragma unroll
    for (int r = 0; r < 8; ++r) {
        int orow = rbase + r;
        if (orow < N) Y[(size_t)orow * OUTC + colbase + m] = acc[r] * norm[orow];
    }
}

// ----------------------------------------------- edge gather + scatter-add (segment_sum)
// One thread per (edge, 4 features): one b128 gather + one index read feeds
// four non-returning global_atomic_add_f32 (atomics are 4B-granular anyway).
template <int F>
__global__ void scatter_add(const float* __restrict__ H, const int* __restrict__ src,
                            const int* __restrict__ dst, float* __restrict__ AGG, int E) {
    constexpr int FV = F / 4;
    long long i = (long long)blockIdx.x * blockDim.x + threadIdx.x;
    if (i >= (long long)E * FV) return;
    int e = (int)(i / FV);
    int f = (int)(i % FV) * 4;
    int sidx = src[e];
    int didx = dst[e];
    const float4 h = *(const float4*)(H + (size_t)sidx * F + f);
    float* a = AGG + (size_t)didx * F + f;
    unsafeAtomicAdd(a + 0, h.x);
    unsafeAtomicAdd(a + 1, h.y);
    unsafeAtomicAdd(a + 2, h.z);
    unsafeAtomicAdd(a + 3, h.w);
}

// ----------------------------------------------- in-place: A = (A * in_norm + b) [relu]
template <int F, bool RELU>
__global__ void epilogue(float* __restrict__ A, const float* __restrict__ in_norm,
                         const float* __restrict__ b, int N) {
    constexpr int FV = F / 4;
    long long i = (long long)blockIdx.x * blockDim.x + threadIdx.x;
    if (i >= (long long)N * FV) return;
    int row = (int)(i / FV);
    int f   = (int)(i % FV) * 4;
    float s = in_norm[row];
    float4 v = *(float4*)(A + (size_t)row * F + f);
    const float4 bb = *(const float4*)(b + f);
    v.x = v.x * s + bb.x;
    v.y = v.y * s + bb.y;
    v.z = v.z * s + bb.z;
    v.w = v.w * s + bb.w;
    if (RELU) {
        v.x = fmaxf(v.x, 0.0f);
        v.y = fmaxf(v.y, 0.0f);
        v.z = fmaxf(v.z, 0.0f);
        v.w = fmaxf(v.w, 0.0f);
    }
    *(float4*)(A + (size_t)row * F + f) = v;
}

// --------------------------------------------------------------------------- launch
extern "C" void kernel_launch(void* const* d_in, const int* in_sizes, int n_in,
                              void* d_out, int out_size, void* d_ws, size_t ws_size,
                              hipStream_t stream) {
    const float* X  = (const float*)d_in[0];   // [N,128]
    const float* W1 = (const float*)d_in[1];   // [128,64]
    const float* b1 = (const float*)d_in[2];   // [64]
    const float* W2 = (const float*)d_in[3];   // [64,32]
    const float* b2 = (const float*)d_in[4];   // [32]
    const int*   src = (const int*)d_in[5];    // [E]
    const int*   dst = (const int*)d_in[6];    // [E]
    const int N = in_sizes[0] / 128;
    const int E = in_sizes[5];
    float* out = (float*)d_out;                // [N,32]

    // workspace layout (floats): deg(2N) | norms(2N) | H(64N, reused for layer2) | AGG1(64N)
    float* ws     = (float*)d_ws;
    float* outdeg = ws;                 ws += N;
    float* indeg  = ws;                 ws += N;
    float* onorm  = ws;                 ws += N;
    float* inorm  = ws;                 ws += N;
    float* H      = ws;                 ws += (size_t)N * 64;   // GEMM output (both layers)
    float* AGG1   = ws;                 ws += (size_t)N * 64;   // layer-1 aggregation

    // zero accumulators (harness poisons buffers; we must own init every call)
    zero_f32<<<1024, 256, 0, stream>>>(outdeg, 2LL * N);
    zero_f32<<<2048, 256, 0, stream>>>(AGG1, (long long)N * 64);
    zero_f32<<<2048, 256, 0, stream>>>(out, (long long)N * 32);

    // degrees + norms
    degree_kernel<<<(E + 255) / 256, 256, 0, stream>>>(src, dst, outdeg, indeg, E);
    norm_kernel<<<(N + 255) / 256, 256, 0, stream>>>(outdeg, indeg, onorm, inorm, N);

    // layer 1: H = onorm * (X @ W1) ; AGG1 = segsum ; AGG1 = relu(AGG1*inorm + b1)
    gemm_wmma<128, 64><<<(N + 31) / 32, 256, 0, stream>>>(X, W1, onorm, H, N);
    {
        long long tot = (long long)E * 16;
        scatter_add<64><<<(int)((tot + 255) / 256), 256, 0, stream>>>(H, src, dst, AGG1, E);
    }
    {
        long long tot = (long long)N * 16;
        epilogue<64, true><<<(int)((tot + 255) / 256), 256, 0, stream>>>(AGG1, inorm, b1, N);
    }

    // layer 2: H = onorm * (AGG1 @ W2) ; out = segsum ; out = out*inorm + b2
    gemm_wmma<64, 32><<<(N + 63) / 64, 256, 0, stream>>>(AGG1, W2, onorm, H, N);
    {
        long long tot = (long long)E * 8;
        scatter_add<32><<<(int)((tot + 255) / 256), 256, 0, stream>>>(H, src, dst, out, E);
    }
    {
        long long tot = (long long)N * 8;
        epilogue<32, false><<<(int)((tot + 255) / 256), 256, 0, stream>>>(out, inorm, b2, N);
    }
}